// BaselineGNN_9577777070497
// MI455X (gfx1250) — compile-verified
//
#include <hip/hip_runtime.h>
#include <math.h>

#define NN 20000
#define NE 640000
#define ETOT (NE + NN)          // edges + self loops
#define NG 64
#define NEG_SLOPE 0.2f

typedef float v2f __attribute__((ext_vector_type(2)));
typedef float v8f __attribute__((ext_vector_type(8)));

// ---------------- utility fills ----------------
__global__ void fill_f32_k(float* __restrict__ p, long n, float v) {
    long i = (long)blockIdx.x * blockDim.x + threadIdx.x;
    if (i < n) p[i] = v;
}

// ---------------- fp32 WMMA GEMM: C[nrows,ncols] = A[nrows,K] @ B[K,ncols] ----------------
// One wave per 16x16 output tile. nrows % 16 == 0 guaranteed (20000 = 1250*16),
// so EXEC is all-ones inside every wave that runs WMMA.
__global__ void gemm_wmma_f32_k(const float* __restrict__ A, const float* __restrict__ B,
                                float* __restrict__ C, int nrows, int K, int ncols) {
    const int lane = threadIdx.x & 31;
    const int wid  = (blockIdx.x * blockDim.x + threadIdx.x) >> 5;
    const int tiles_n = ncols >> 4;
    const int tiles_m = nrows >> 4;
    const int mt = wid / tiles_n;
    const int nt = wid - mt * tiles_n;
    if (mt >= tiles_m) return;                       // wave-uniform guard

    const int row0 = mt << 4;
    const int m    = lane & 15;
    const int kh   = (lane >> 4) << 1;               // 0 or 2 (K sub-slot)
    const int col  = (nt << 4) + (lane & 15);

    const float* __restrict__ arow = A + (long)(row0 + m) * K + kh;

    v8f acc = {0.f, 0.f, 0.f, 0.f, 0.f, 0.f, 0.f, 0.f};
    #pragma unroll 4
    for (int k = 0; k < K; k += 4) {
        v2f a, b;
        a[0] = arow[k];                              // A[row0+m][k+kh]
        a[1] = arow[k + 1];                          // A[row0+m][k+kh+1]
        b[0] = B[(long)(k + kh)     * ncols + col];  // B[k+kh][col]
        b[1] = B[(long)(k + kh + 1) * ncols + col];  // B[k+kh+1][col]
        acc = __builtin_amdgcn_wmma_f32_16x16x4_f32(false, a, false, b,
                                                    (short)0, acc, false, false);
    }
    const int rbase = row0 + ((lane >> 4) << 3);     // M = i + 8*(lane>=16)
    #pragma unroll
    for (int i = 0; i < 8; ++i)
        C[(long)(rbase + i) * ncols + col] = acc[i];
}

// ---------------- attention logits: al_s/al_d[n,h] = <xh[n,h,:], a_*[h,:]> ----------------
__global__ void att_logits_k(const float* __restrict__ xh,
                             const float* __restrict__ a_src, const float* __restrict__ a_dst,
                             float* __restrict__ als, float* __restrict__ ald,
                             int n_nodes, int heads, int ch) {
    int tid = blockIdx.x * blockDim.x + threadIdx.x;
    if (tid >= n_nodes * heads) return;
    int node = tid / heads, h = tid - node * heads;
    const float* base = xh + (long)node * heads * ch + (long)h * ch;
    const float* as = a_src + (long)h * ch;
    const float* ad = a_dst + (long)h * ch;
    float s = 0.f, d = 0.f;
    #pragma unroll 8
    for (int c = 0; c < 64; ++c) { s += base[c] * as[c]; d += base[c] * ad[c]; }
    als[tid] = s;
    ald[tid] = d;
}

// ---------------- float <-> monotone-uint encoding for atomicMax ----------------
__device__ __forceinline__ unsigned fenc(float f) {
    unsigned u = __float_as_uint(f);
    return (u & 0x80000000u) ? ~u : (u | 0x80000000u);
}
__device__ __forceinline__ float fdec(unsigned u) {
    unsigned b = (u & 0x80000000u) ? (u & 0x7fffffffu) : ~u;
    return __uint_as_float(b);
}

__device__ __forceinline__ void get_edge(const int* __restrict__ ei, int e, int& s, int& d) {
    if (e < NE) { s = ei[e]; d = ei[NE + e]; }
    else        { s = d = e - NE; }                  // appended self loop
}
__device__ __forceinline__ float leaky(float x) { return x > 0.f ? x : NEG_SLOPE * x; }

// ---------------- edge pass 1: segment max of leaky(e) over dst ----------------
__global__ void edge_max_k(const int* __restrict__ ei,
                           const float* __restrict__ als, const float* __restrict__ ald,
                           unsigned* __restrict__ mx, int heads) {
    int tid = blockIdx.x * blockDim.x + threadIdx.x;
    if (tid >= ETOT * heads) return;
    int e = tid / heads, h = tid - e * heads;
    int s, d; get_edge(ei, e, s, d);
    float el = leaky(als[s * heads + h] + ald[d * heads + h]);
    atomicMax(&mx[d * heads + h], fenc(el));
}

// ---------------- edge pass 2: ex = exp(e - max[dst]); z[dst] += ex ----------------
__global__ void edge_exp_k(const int* __restrict__ ei,
                           const float* __restrict__ als, const float* __restrict__ ald,
                           const unsigned* __restrict__ mx, float* __restrict__ z,
                           float* __restrict__ exbuf, int heads) {
    int tid = blockIdx.x * blockDim.x + threadIdx.x;
    if (tid >= ETOT * heads) return;
    int e = tid / heads, h = tid - e * heads;
    int s, d; get_edge(ei, e, s, d);
    float el = leaky(als[s * heads + h] + ald[d * heads + h]);
    float ex = expf(el - fdec(mx[d * heads + h]));
    exbuf[(long)e * heads + h] = ex;
    atomicAdd(&z[d * heads + h], ex);
}

// ---------------- edge pass 3: agg[dst] += alpha * xh[src], 4 channels/thread ----------------
__global__ void edge_agg_k(const int* __restrict__ ei, const float* __restrict__ xh,
                           const float* __restrict__ exbuf, const float* __restrict__ z,
                           float* __restrict__ agg, int heads, int ch) {
    int hc = heads * ch;
    int chunks = hc >> 2;
    long tid = (long)blockIdx.x * blockDim.x + threadIdx.x;
    if (tid >= (long)ETOT * chunks) return;
    int e  = (int)(tid / chunks);
    int q  = (int)(tid - (long)e * chunks);
    int c0 = q << 2;
    int h  = c0 / ch;
    int s, d; get_edge(ei, e, s, d);
    float alpha = exbuf[(long)e * heads + h] / z[d * heads + h];
    const float4 v = *(const float4*)(xh + (long)s * hc + c0);
    float* o = agg + (long)d * hc + c0;
    atomicAdd(o + 0, v.x * alpha);
    atomicAdd(o + 1, v.y * alpha);
    atomicAdd(o + 2, v.z * alpha);
    atomicAdd(o + 3, v.w * alpha);
}

// ---------------- bias (+ optional relu), in-place capable ----------------
__global__ void bias_act_k(const float* __restrict__ in, const float* __restrict__ bias,
                           float* __restrict__ out, long total, int hc, int do_relu) {
    long i = (long)blockIdx.x * blockDim.x + threadIdx.x;
    if (i >= total) return;
    int c = (int)(i % hc);
    float v = in[i] + bias[c];
    out[i] = do_relu ? fmaxf(v, 0.f) : v;
}

// ---------------- global add pool into d_out[0..4095] ----------------
__global__ void pool_k(const float* __restrict__ h2, const int* __restrict__ batch,
                       float* __restrict__ ge) {
    long i = (long)blockIdx.x * blockDim.x + threadIdx.x;
    if (i >= (long)NN * 64) return;
    int n = (int)(i >> 6), c = (int)(i & 63);
    atomicAdd(&ge[batch[n] * 64 + c], h2[i]);
}

// ---------------- classifier + log-softmax + loss + probs (single block) ----------------
__global__ void classifier_k(const float* __restrict__ ge,
                             const float* __restrict__ Wc1, const float* __restrict__ bc1,
                             const float* __restrict__ Wc2, const float* __restrict__ bc2,
                             const int* __restrict__ y, float* __restrict__ out) {
    __shared__ float hid[64 * 64];
    __shared__ float logits[64 * 2];
    __shared__ float lpy[64];
    __shared__ float corr[64];
    int t = threadIdx.x;                             // 256 threads
    for (int i = t; i < 64 * 64; i += 256) {
        int g = i >> 6, j = i & 63;
        float s = bc1[j];
        #pragma unroll 8
        for (int c = 0; c < 64; ++c) s += ge[g * 64 + c] * Wc1[c * 64 + j];
        hid[i] = fmaxf(s, 0.f);
    }
    __syncthreads();
    for (int i = t; i < 128; i += 256) {
        int g = i >> 1, k = i & 1;
        float s = bc2[k];
        #pragma unroll 8
        for (int j = 0; j < 64; ++j) s += hid[g * 64 + j] * Wc2[j * 2 + k];
        logits[i] = s;
    }
    __syncthreads();
    if (t < 64) {
        int g = t;
        float l0 = logits[g * 2], l1 = logits[g * 2 + 1];
        float m  = fmaxf(l0, l1);
        float lse = m + logf(expf(l0 - m) + expf(l1 - m));
        float lp0 = l0 - lse, lp1 = l1 - lse;
        out[4098 + g * 2]     = expf(lp0);           // probs
        out[4098 + g * 2 + 1] = expf(lp1);
        int yi = y[g];
        lpy[g]  = yi ? lp1 : lp0;
        int pred = (lp1 > lp0) ? 1 : 0;              // ties -> 0, matches argmax
        corr[g] = (pred == yi) ? 1.f : 0.f;
    }
    __syncthreads();
    if (t == 0) {
        float sl = 0.f, sc = 0.f;
        for (int g = 0; g < 64; ++g) { sl += lpy[g]; sc += corr[g]; }
        out[4096] = -sl / 64.f;                      // cls_loss
        out[4097] = sc;                              // correct_num
    }
}

static inline int nblk(long n, int bs) { return (int)((n + bs - 1) / bs); }

extern "C" void kernel_launch(void* const* d_in, const int* in_sizes, int n_in,
                              void* d_out, int out_size, void* d_ws, size_t ws_size,
                              hipStream_t stream) {
    // inputs per setup_inputs() order
    const float* x      = (const float*)d_in[0];
    const int*   ei     = (const int*)d_in[1];      // [2, NE]
    const int*   batch  = (const int*)d_in[2];
    const int*   y      = (const int*)d_in[3];
    const float* W1     = (const float*)d_in[4];
    const float* a_src1 = (const float*)d_in[5];
    const float* a_dst1 = (const float*)d_in[6];
    const float* b1     = (const float*)d_in[7];
    const float* W2     = (const float*)d_in[8];
    const float* a_src2 = (const float*)d_in[9];
    const float* a_dst2 = (const float*)d_in[10];
    const float* b2     = (const float*)d_in[11];
    const float* Wc1    = (const float*)d_in[12];
    const float* bc1    = (const float*)d_in[13];
    const float* Wc2    = (const float*)d_in[14];
    const float* bc2    = (const float*)d_in[15];
    float* out = (float*)d_out;

    // workspace carve-up (floats)
    float* W = (float*)d_ws;
    long o = 0;
    float*    bufA = W + o; o += (long)NN * 128;     // xh1, later xh2
    float*    bufB = W + o; o += (long)NN * 128;     // agg1/h1, later agg2/h2
    float*    als1 = W + o; o += (long)NN * 2;
    float*    ald1 = W + o; o += (long)NN * 2;
    unsigned* mx1  = (unsigned*)(W + o); o += (long)NN * 2;
    float*    z1   = W + o; o += (long)NN * 2;
    float*    als2 = W + o; o += NN;
    float*    ald2 = W + o; o += NN;
    unsigned* mx2  = (unsigned*)(W + o); o += NN;
    float*    z2   = W + o; o += NN;
    float*    ex1  = W + o; o += (long)ETOT * 2;
    float*    ex2  = W + o; o += (long)ETOT;

    const int BS = 256;

    // ---- init (fenc maps every float to > 0, so uint 0 == -inf for the max bufs) ----
    fill_f32_k<<<nblk(4096, BS), BS, 0, stream>>>(out, 4096, 0.f);
    fill_f32_k<<<nblk((long)NN * 2, BS), BS, 0, stream>>>((float*)mx1, (long)NN * 2, 0.f);
    fill_f32_k<<<nblk((long)NN * 2, BS), BS, 0, stream>>>(z1, (long)NN * 2, 0.f);
    fill_f32_k<<<nblk(NN, BS), BS, 0, stream>>>((float*)mx2, NN, 0.f);
    fill_f32_k<<<nblk(NN, BS), BS, 0, stream>>>(z2, NN, 0.f);
    fill_f32_k<<<nblk((long)NN * 128, BS), BS, 0, stream>>>(bufB, (long)NN * 128, 0.f);

    // ---- GAT layer 1 (heads=2, ch=64) ----
    {   // xh1 = x @ W1   [20000,128] x [128,128]
        long waves = (long)(NN / 16) * (128 / 16);
        gemm_wmma_f32_k<<<nblk(waves * 32, BS), BS, 0, stream>>>(x, W1, bufA, NN, 128, 128);
    }
    att_logits_k<<<nblk((long)NN * 2, BS), BS, 0, stream>>>(bufA, a_src1, a_dst1, als1, ald1, NN, 2, 64);
    edge_max_k<<<nblk((long)ETOT * 2, BS), BS, 0, stream>>>(ei, als1, ald1, mx1, 2);
    edge_exp_k<<<nblk((long)ETOT * 2, BS), BS, 0, stream>>>(ei, als1, ald1, mx1, z1, ex1, 2);
    edge_agg_k<<<nblk((long)ETOT * 32, BS), BS, 0, stream>>>(ei, bufA, ex1, z1, bufB, 2, 64);
    bias_act_k<<<nblk((long)NN * 128, BS), BS, 0, stream>>>(bufB, b1, bufB, (long)NN * 128, 128, 1); // h1 = relu(.)

    // ---- GAT layer 2 (heads=1, ch=64) ----
    {   // xh2 = h1 @ W2   [20000,128] x [128,64]  -> bufA (xh1 dead)
        long waves = (long)(NN / 16) * (64 / 16);
        gemm_wmma_f32_k<<<nblk(waves * 32, BS), BS, 0, stream>>>(bufB, W2, bufA, NN, 128, 64);
    }
    att_logits_k<<<nblk(NN, BS), BS, 0, stream>>>(bufA, a_src2, a_dst2, als2, ald2, NN, 1, 64);
    fill_f32_k<<<nblk((long)NN * 64, BS), BS, 0, stream>>>(bufB, (long)NN * 64, 0.f);  // agg2 (h1 dead)
    edge_max_k<<<nblk(ETOT, BS), BS, 0, stream>>>(ei, als2, ald2, mx2, 1);
    edge_exp_k<<<nblk(ETOT, BS), BS, 0, stream>>>(ei, als2, ald2, mx2, z2, ex2, 1);
    edge_agg_k<<<nblk((long)ETOT * 16, BS), BS, 0, stream>>>(ei, bufA, ex2, z2, bufB, 1, 64);
    bias_act_k<<<nblk((long)NN * 64, BS), BS, 0, stream>>>(bufB, b2, bufB, (long)NN * 64, 64, 0);    // h2

    // ---- pooling + classifier + loss ----
    pool_k<<<nblk((long)NN * 64, BS), BS, 0, stream>>>(bufB, batch, out);
    classifier_k<<<1, 256, 0, stream>>>(out, Wc1, bc1, Wc2, bc2, y, out);
    (void)in_sizes; (void)n_in; (void)out_size; (void)ws_size;
}